// Model_41068477285087
// MI455X (gfx1250) — compile-verified
//
#include <hip/hip_runtime.h>

typedef __attribute__((ext_vector_type(2))) float v2f;
typedef __attribute__((ext_vector_type(8))) float v8f;

// ---------------- elementwise utility kernels ----------------
__global__ void k_zero_f4(float4* __restrict__ dst, long n4) {
  long i = (long)blockIdx.x * blockDim.x + threadIdx.x;
  if (i < n4) dst[i] = make_float4(0.f, 0.f, 0.f, 0.f);
}

__global__ void k_copy_f4(const float4* __restrict__ src, float4* __restrict__ dst, long n4) {
  long i = (long)blockIdx.x * blockDim.x + threadIdx.x;
  if (i < n4) dst[i] = src[i];
}

// ---------------- COO SpMM, L2-resident f32 atomics ----------------
// 16 lanes cooperate on one nonzero: each lane moves one float4 (16B) of the
// 64-float (256B) embedding row -> one coalesced 256B burst per edge.
// Accumulation via HW global_atomic_add_f32; src+dst tables (~154MB) fit the
// 192MB L2, so atomics and gathers stay on-chip.
__global__ void k_spmm_coo(const int* __restrict__ rows, const int* __restrict__ cols,
                           const float* __restrict__ vals, const float* __restrict__ x,
                           float* __restrict__ out, int nnz) {
  long t = (long)blockIdx.x * blockDim.x + threadIdx.x;
  int edge = (int)(t >> 4);
  if (edge >= nnz) return;
  int sub = (int)(t & 15);
  int r = rows[edge];
  int c = cols[edge];
  float v = vals[edge];
  float4 xv = ((const float4*)(x + (long)c * 64))[sub];
  float* o = out + (long)r * 64 + sub * 4;
  unsafeAtomicAdd(o + 0, v * xv.x);
  unsafeAtomicAdd(o + 1, v * xv.y);
  unsafeAtomicAdd(o + 2, v * xv.z);
  unsafeAtomicAdd(o + 3, v * xv.w);
}

// ---------------- 3-layer mean via V_WMMA_F32_16X16X4_F32 ----------------
// mean = (1/3,1/3,1/3) . [E0;E1;E2]  (rank-1 contraction over layer axis)
// A(16x4): lane m   -> {A[m][0],A[m][1]} = {1/3,1/3}
//          lane 16+m-> {A[m][2],A[m][3]} = {1/3, 0 }
// B(4x16): lane n   -> {B[0][n],B[1][n]} = {E0[b+n],E1[b+n]}
//          lane 16+n-> {B[2][n],B[3][n]} = {E2[b+n], 0     }
// D[0][n] = (E0+E1+E2)/3 -> VGPR0 of lanes 0..15. In-place dst==e0 is safe:
// each index is read+written by the same wave in program order.
// EXEC stays full through the WMMA (no early return; tail waves clamp base
// and just suppress the store).
__global__ void k_mean3_wmma(const float* __restrict__ e0, const float* __restrict__ e1,
                             const float* __restrict__ e2, float* __restrict__ dst,
                             long nelem) {
  const int lane = threadIdx.x & 31;
  const int n = lane & 15;
  const int hi = lane >> 4;
  long wave = (long)blockIdx.x * (blockDim.x >> 5) + (threadIdx.x >> 5);
  long base = wave * 16;
  bool valid = (base + 16) <= nelem;
  long b = valid ? base : 0;

  const float third = 1.0f / 3.0f;
  v2f a;
  a[0] = third;
  a[1] = hi ? 0.0f : third;
  v2f bm;
  bm[0] = hi ? e2[b + n] : e0[b + n];
  bm[1] = hi ? 0.0f : e1[b + n];
  v8f c = {};
  c = __builtin_amdgcn_wmma_f32_16x16x4_f32(false, a, false, bm, (short)0, c, false, false);
  if (valid && hi == 0) dst[b + n] = c[0];
}

extern "C" void kernel_launch(void* const* d_in, const int* in_sizes, int n_in,
                              void* d_out, int out_size, void* d_ws, size_t ws_size,
                              hipStream_t stream) {
  (void)n_in; (void)out_size; (void)d_ws; (void)ws_size;

  const int*   adj_rows = (const int*)d_in[0];
  const int*   adj_cols = (const int*)d_in[1];
  const float* adj_vals = (const float*)d_in[2];
  const int*   d_rows   = (const int*)d_in[3];
  const int*   d_cols   = (const int*)d_in[4];
  const float* d_vals   = (const float*)d_in[5];
  const float* uE       = (const float*)d_in[6];
  const float* iE       = (const float*)d_in[7];

  const int  nnz_adj = in_sizes[0];
  const int  nnz_d   = in_sizes[3];
  const long user64  = (long)in_sizes[6];   // USER * 64 floats
  const long item64  = (long)in_sizes[7];   // ITEM * 64 floats
  const long N64     = user64 + item64;     // N * 64 floats

  // d_out regions: A = final embeds, B = condition embeds, C = u/i copies
  float* A = (float*)d_out;
  float* B = A + N64;
  float* C = B + N64;

  const int THR = 256;
  auto gblk = [](long work, int thr) -> unsigned { return (unsigned)((work + thr - 1) / thr); };

  // 1. E0 = concat(u, i) -> A
  k_copy_f4<<<gblk(user64 / 4, THR), THR, 0, stream>>>((const float4*)uE, (float4*)A, user64 / 4);
  k_copy_f4<<<gblk(item64 / 4, THR), THR, 0, stream>>>((const float4*)iE, (float4*)(A + user64), item64 / 4);

  // 2. E1 = adj @ E0 -> C  (C reused as scratch; rewritten with u/i at the end)
  k_zero_f4<<<gblk(N64 / 4, THR), THR, 0, stream>>>((float4*)C, N64 / 4);
  k_spmm_coo<<<gblk((long)nnz_adj * 16, THR), THR, 0, stream>>>(
      adj_rows, adj_cols, adj_vals, A, C, nnz_adj);

  // 3. E2 = adj @ E1 -> B
  k_zero_f4<<<gblk(N64 / 4, THR), THR, 0, stream>>>((float4*)B, N64 / 4);
  k_spmm_coo<<<gblk((long)nnz_adj * 16, THR), THR, 0, stream>>>(
      adj_rows, adj_cols, adj_vals, C, B, nnz_adj);

  // 4. embeds = mean(E0, E1, E2) -> A (in place over E0), one 16-chunk per wave
  {
    long chunks = N64 / 16;
    int wavesPerBlock = THR / 32;
    unsigned blocks = (unsigned)((chunks + wavesPerBlock - 1) / wavesPerBlock);
    k_mean3_wmma<<<blocks, THR, 0, stream>>>(A, C, B, A, N64);
  }

  // 5. cond = d @ embeds -> B
  k_zero_f4<<<gblk(N64 / 4, THR), THR, 0, stream>>>((float4*)B, N64 / 4);
  k_spmm_coo<<<gblk((long)nnz_d * 16, THR), THR, 0, stream>>>(
      d_rows, d_cols, d_vals, A, B, nnz_d);

  // 6. passthrough copies: u -> C[0:user64), i -> C[user64:N64)
  k_copy_f4<<<gblk(user64 / 4, THR), THR, 0, stream>>>((const float4*)uE, (float4*)C, user64 / 4);
  k_copy_f4<<<gblk(item64 / 4, THR), THR, 0, stream>>>((const float4*)iE, (float4*)(C + user64), item64 / 4);
}